// GPWorldModel_74620761800916
// MI455X (gfx1250) — compile-verified
//
#include <hip/hip_runtime.h>
#include <hip/hip_bf16.h>
#include <math.h>

typedef __attribute__((ext_vector_type(2))) float v2f;
typedef __attribute__((ext_vector_type(8))) float v8f;

// Problem constants
constexpr int NB = 16;   // batch
constexpr int ND = 8;    // obs dims
constexpr int NF = 2;    // action dims
constexpr int NN = 512;  // GP training points
constexpr int NE = ND + NF;  // 10
constexpr int NEP = 12;      // padded K (3 chained K=4 f32 WMMAs)

// ---- workspace layout (floats) ----
constexpr size_t OFF_SIG   = 0;                           // B*E*E      = 1600
constexpr size_t OFF_MU    = OFF_SIG  + NB*NE*NE;         // B*E        = 160
constexpr size_t OFF_AINV  = OFF_MU   + NB*NE;            // B*D*E*E    = 12800
constexpr size_t OFF_LDA   = OFF_AINV + NB*ND*NE*NE;      // B*D        = 128
constexpr size_t OFF_SMAT  = OFF_LDA  + NB*ND;            // D*B*D*E*E  = 102400
constexpr size_t OFF_LDR   = OFF_SMAT + ND*NB*ND*NE*NE;   // D*B*D      = 1024
constexpr size_t OFF_LOGK  = OFF_LDR  + ND*NB*ND;         // B*D*N      = 65536
constexpr size_t OFF_MUD   = OFF_LOGK + NB*ND*NN;         // B*D        = 128
constexpr size_t OFF_V     = OFF_MUD  + NB*ND;            // B*D*E      = 1280
constexpr size_t OFF_EDD   = OFF_V    + NB*ND*NE;         // B*D*D      = 1024
constexpr size_t OFF_TR    = OFF_EDD  + NB*ND*ND;         // B*D        = 128
// total ~185 KF ~ 740 KB

// ================= kernel 1: assemble mu [B,E], Sigma [B,E,E] =================
__global__ void k_assemble(const float* __restrict__ obs_mean,
                           const float* __restrict__ obs_var,
                           const float* __restrict__ act_mean,
                           const float* __restrict__ act_var,
                           const float* __restrict__ cross,
                           float* __restrict__ ws) {
    float* SIG = ws + OFF_SIG;
    float* MU  = ws + OFF_MU;
    for (int idx = threadIdx.x; idx < NB * NE; idx += blockDim.x) {
        int b = idx / NE, e = idx % NE;
        MU[idx] = (e < ND) ? obs_mean[b * ND + e] : act_mean[b * NF + (e - ND)];
    }
    for (int idx = threadIdx.x; idx < NB * NE * NE; idx += blockDim.x) {
        int b = idx / (NE * NE), r = idx % (NE * NE);
        int e = r / NE, f = r % NE;
        float v;
        if (e < ND && f < ND)       v = obs_var[(b * ND + e) * ND + f];
        else if (e < ND)            v = cross[(b * ND + e) * NF + (f - ND)];
        else if (f < ND)            v = cross[(b * ND + f) * NF + (e - ND)];
        else                        v = act_var[(b * NF + (e - ND)) * NF + (f - ND)];
        SIG[idx] = v;
    }
}

// ============ kernel 2: Ainv[b,d] = (Sigma[b]+diag(ell2[d]))^-1, logdetA ============
// one wave (32 lanes) per matrix; lane c owns augmented column c of [A | I]
__global__ void k_ainv(const float* __restrict__ ell, float* __restrict__ ws) {
    int blk = blockIdx.x;           // b*ND + d
    int b = blk / ND, d = blk % ND;
    int c = threadIdx.x;            // column id (0..19 meaningful)
    const float* SIG = ws + OFF_SIG + (size_t)b * NE * NE;
    float m[NE];
#pragma unroll
    for (int i = 0; i < NE; i++) {
        float v;
        if (c < NE) {
            float l = ell[d * NE + c];
            v = SIG[i * NE + c] + ((i == c) ? l * l : 0.f);
        } else {
            v = (i == c - NE) ? 1.f : 0.f;
        }
        m[i] = v;
    }
    float ld = 0.f;
#pragma unroll
    for (int k = 0; k < NE; k++) {
        float piv = __shfl(m[k], k, 32);
        ld += __logf(piv);
        float inv = 1.f / piv;
        m[k] *= inv;
#pragma unroll
        for (int i = 0; i < NE; i++) {
            if (i == k) continue;
            float f = __shfl(m[i], k, 32);
            m[i] -= f * m[k];
        }
    }
    float* AINV = ws + OFF_AINV + (size_t)blk * NE * NE;
    if (c >= NE && c < 2 * NE) {
#pragma unroll
        for (int i = 0; i < NE; i++) AINV[i * NE + (c - NE)] = m[i];
    }
    if (c == 0) ws[OFF_LDA + blk] = ld;
}

// ====== kernel 3: S[a,b,d] = R^-1 Sigma, R = Sigma .* col(iLam[a]+iLam[d]) + I ======
__global__ void k_ssolve(const float* __restrict__ ell, float* __restrict__ ws) {
    int blk = blockIdx.x;                 // ((a*NB)+b)*ND + d
    int a = blk / (NB * ND);
    int b = (blk / ND) % NB;
    int d = blk % ND;
    int c = threadIdx.x;
    const float* SIG = ws + OFF_SIG + (size_t)b * NE * NE;
    float m[NE];
#pragma unroll
    for (int i = 0; i < NE; i++) {
        float v;
        if (c < NE) {
            float la = ell[a * NE + c], ldd = ell[d * NE + c];
            float ils = 1.f / (la * la) + 1.f / (ldd * ldd);
            v = SIG[i * NE + c] * ils + ((i == c) ? 1.f : 0.f);
        } else if (c < 2 * NE) {
            v = SIG[i * NE + (c - NE)];
        } else {
            v = 0.f;
        }
        m[i] = v;
    }
    float ld = 0.f;
#pragma unroll
    for (int k = 0; k < NE; k++) {
        float piv = __shfl(m[k], k, 32);
        ld += __logf(piv);
        float inv = 1.f / piv;
        m[k] *= inv;
#pragma unroll
        for (int i = 0; i < NE; i++) {
            if (i == k) continue;
            float f = __shfl(m[i], k, 32);
            m[i] -= f * m[k];
        }
    }
    float* S = ws + OFF_SMAT + (size_t)blk * NE * NE;
    if (c >= NE && c < 2 * NE) {
#pragma unroll
        for (int i = 0; i < NE; i++) S[i * NE + (c - NE)] = m[i];
    }
    if (c == 0) ws[OFF_LDR + blk] = ld;
}

__device__ __forceinline__ float blockReduce256(float v, float* red) {
    int tid = threadIdx.x;
    red[tid] = v;
    __syncthreads();
    for (int s = 128; s > 0; s >>= 1) {
        if (tid < s) red[tid] += red[tid + s];
        __syncthreads();
    }
    float r = red[0];
    __syncthreads();
    return r;
}

// ====== kernel 4: quad/q/logk; mu_delta, w, V per (b,d) ======
__global__ void k_qkern(const float* __restrict__ X,
                        const float* __restrict__ ell,
                        const float* __restrict__ alpha_sq,
                        const float* __restrict__ beta,
                        float* __restrict__ ws) {
    __shared__ float sAinv[NE * NE];
    __shared__ float sSig[NE * NE];
    __shared__ float sMu[NE];
    __shared__ float sIl[NE];
    __shared__ float red[256];
    __shared__ float sW[NE];
    __shared__ float sAw[NE];

    int b = blockIdx.x / ND, d = blockIdx.x % ND;
    int tid = threadIdx.x;
    if (tid < NE * NE) {
        sAinv[tid] = ws[OFF_AINV + (size_t)(b * ND + d) * NE * NE + tid];
        sSig[tid]  = ws[OFF_SIG + (size_t)b * NE * NE + tid];
    }
    if (tid < NE) {
        sMu[tid] = ws[OFF_MU + b * NE + tid];
        float l = ell[d * NE + tid];
        sIl[tid] = 1.f / (l * l);
    }
    __syncthreads();

    float ldl = 0.f;
#pragma unroll
    for (int e = 0; e < NE; e++) {
        float l = ell[d * NE + e];
        ldl += __logf(l * l);
    }
    float lda = ws[OFF_LDA + b * ND + d];
    float a2 = alpha_sq[d];
    float la2 = __logf(a2);
    float coef = a2 * __expf(0.5f * (ldl - lda));

    float md = 0.f;
    float wp[NE];
#pragma unroll
    for (int e = 0; e < NE; e++) wp[e] = 0.f;

    for (int n = tid; n < NN; n += 256) {
        float nu[NE];
#pragma unroll
        for (int e = 0; e < NE; e++) nu[e] = X[n * NE + e] - sMu[e];
        float quad = 0.f, ss = 0.f;
#pragma unroll
        for (int e = 0; e < NE; e++) {
            float t = 0.f;
#pragma unroll
            for (int f = 0; f < NE; f++) t += sAinv[e * NE + f] * nu[f];
            quad += nu[e] * t;
            ss += nu[e] * nu[e] * sIl[e];
        }
        float qv = coef * __expf(-0.5f * quad);
        ws[OFF_LOGK + (size_t)(b * ND + d) * NN + n] = la2 - 0.5f * ss;
        float bt = beta[d * NN + n];
        md += bt * qv;
#pragma unroll
        for (int e = 0; e < NE; e++) wp[e] += bt * qv * nu[e];
    }

    float mdt = blockReduce256(md, red);
    if (tid == 0) ws[OFF_MUD + b * ND + d] = mdt;
    for (int e = 0; e < NE; e++) {
        float we = blockReduce256(wp[e], red);
        if (tid == 0) sW[e] = we;
    }
    __syncthreads();
    if (tid < NE) {
        float t = 0.f;
#pragma unroll
        for (int g = 0; g < NE; g++) t += sAinv[tid * NE + g] * sW[g];
        sAw[tid] = t;
    }
    __syncthreads();
    if (tid < NE) {
        float t = 0.f;
#pragma unroll
        for (int f = 0; f < NE; f++) t += sSig[tid * NE + f] * sAw[f];
        ws[OFF_V + (size_t)(b * ND + d) * NE + tid] = t;
    }
}

// ====== kernel 5 (heavy, WMMA): per (a,b,d) compute EDD[b,a,d] and trace ======
// Q[i,j] = ea[i] * eb[j] * exp(M[i,j]),  M = PaS . Pd^T  (K padded 10->12)
// ea[i] = exp(ca[i] - ld/4), eb[j] = exp(cb[j] - ld/4); wa = beta_a.*ea, wb = beta_d.*eb
__global__ void k_heavy(const float* __restrict__ X,
                        const float* __restrict__ ell,
                        const float* __restrict__ beta,
                        const float* __restrict__ invK,
                        float* __restrict__ ws) {
    __shared__ __align__(16) float sPaS[NN * NEP];  // 24 KB
    __shared__ __align__(16) float sPd[NN * NEP];   // 24 KB
    __shared__ __align__(16) float sWa[NN];
    __shared__ __align__(16) float sWb[NN];
    __shared__ __align__(16) float sEa[NN];
    __shared__ __align__(16) float sEb[NN];
    __shared__ float sS[NE * NE];
    __shared__ float sMu[NE];
    __shared__ float sIlA[NE];
    __shared__ float sIlD[NE];
    __shared__ float sRed[16];

    int g = blockIdx.x;
    int a = g / (NB * ND);
    int b = (g / ND) % NB;
    int d = g % ND;
    int tid = threadIdx.x;

    if (tid < NE * NE)
        sS[tid] = ws[OFF_SMAT + (size_t)g * NE * NE + tid];
    if (tid < NE) {
        sMu[tid] = ws[OFF_MU + b * NE + tid];
        float la = ell[a * NE + tid], ldd = ell[d * NE + tid];
        sIlA[tid] = 1.f / (la * la);
        sIlD[tid] = 1.f / (ldd * ldd);
    }
    __syncthreads();

    const float* logkA = ws + OFF_LOGK + (size_t)(b * ND + a) * NN;
    const float* logkD = ws + OFF_LOGK + (size_t)(b * ND + d) * NN;
    const float qld = 0.25f * ws[OFF_LDR + g];  // ld/4 folded into each side

    // phase 1: build PaS, Pd, ea/eb/wa/wb caches
    for (int n = tid; n < NN; n += 256) {
        float nu[NE], pa[NE], pd[NE];
#pragma unroll
        for (int e = 0; e < NE; e++) {
            nu[e] = X[n * NE + e] - sMu[e];
            pa[e] = nu[e] * sIlA[e];
            pd[e] = nu[e] * sIlD[e];
        }
        float da = 0.f, db = 0.f;
#pragma unroll
        for (int f = 0; f < NE; f++) {
            float pas = 0.f, pds = 0.f;
#pragma unroll
            for (int e = 0; e < NE; e++) {
                pas += pa[e] * sS[e * NE + f];
                pds += pd[e] * sS[e * NE + f];
            }
            sPaS[n * NEP + f] = pas;
            sPd[n * NEP + f] = pd[f];
            da += pas * pa[f];
            db += pds * pd[f];
        }
        sPaS[n * NEP + 10] = 0.f; sPaS[n * NEP + 11] = 0.f;
        sPd[n * NEP + 10] = 0.f;  sPd[n * NEP + 11] = 0.f;
        float eav = __expf(logkA[n] + 0.5f * da - qld);
        float ebv = __expf(logkD[n] + 0.5f * db - qld);
        sEa[n] = eav;
        sEb[n] = ebv;
        sWa[n] = beta[a * NN + n] * eav;
        sWb[n] = beta[d * NN + n] * ebv;
    }
    __syncthreads();

    const float* invKa = invK + (size_t)a * NN * NN;

    int wave = tid >> 5, lane = tid & 31;
    int lhalf = lane >> 4;      // 0: lanes 0-15, 1: lanes 16-31
    int l16 = lane & 15;

    float edd = 0.f, trc = 0.f;

    if (a == d) {
        // diagonal slice: also accumulate trace term sum_ij invK[a][i,j]*Q[i,j]
        for (int ti = wave; ti < 32; ti += 8) {
            int arow = ti * 16 + l16;
            v2f av0, av1, av2;   // A frags cached across the tj sweep
            av0.x = sPaS[arow * NEP + 2 * lhalf];     av0.y = sPaS[arow * NEP + 2 * lhalf + 1];
            av1.x = sPaS[arow * NEP + 4 + 2 * lhalf]; av1.y = sPaS[arow * NEP + 4 + 2 * lhalf + 1];
            av2.x = sPaS[arow * NEP + 8 + 2 * lhalf]; av2.y = sPaS[arow * NEP + 8 + 2 * lhalf + 1];
            int i0 = ti * 16 + 8 * lhalf;            // 8 consecutive i per lane
            float4 wa0 = *(const float4*)&sWa[i0];
            float4 wa1 = *(const float4*)&sWa[i0 + 4];
            float4 ea0 = *(const float4*)&sEa[i0];
            float4 ea1 = *(const float4*)&sEa[i0 + 4];
            float wav[8] = {wa0.x, wa0.y, wa0.z, wa0.w, wa1.x, wa1.y, wa1.z, wa1.w};
            float eav[8] = {ea0.x, ea0.y, ea0.z, ea0.w, ea1.x, ea1.y, ea1.z, ea1.w};
            for (int tj = 0; tj < 32; tj++) {
                int bcol = tj * 16 + l16;
                v2f bv0, bv1, bv2;
                bv0.x = sPd[bcol * NEP + 2 * lhalf];     bv0.y = sPd[bcol * NEP + 2 * lhalf + 1];
                bv1.x = sPd[bcol * NEP + 4 + 2 * lhalf]; bv1.y = sPd[bcol * NEP + 4 + 2 * lhalf + 1];
                bv2.x = sPd[bcol * NEP + 8 + 2 * lhalf]; bv2.y = sPd[bcol * NEP + 8 + 2 * lhalf + 1];
                v8f acc = {};
                acc = __builtin_amdgcn_wmma_f32_16x16x4_f32(false, av0, false, bv0, (short)0, acc, false, false);
                acc = __builtin_amdgcn_wmma_f32_16x16x4_f32(false, av1, false, bv1, (short)0, acc, false, false);
                acc = __builtin_amdgcn_wmma_f32_16x16x4_f32(false, av2, false, bv2, (short)0, acc, false, false);
                // batch the 8 strided invK loads for this tile
                const float* kp = invKa + (size_t)i0 * NN + bcol;
                float kv[8];
#pragma unroll
                for (int r = 0; r < 8; r++) kv[r] = kp[(size_t)r * NN];
                if (tj + 1 < 32) __builtin_prefetch(kp + 16, 0, 3);  // next tile's row
                float wbj = sWb[bcol], ebj = sEb[bcol];
                float se = 0.f, st = 0.f;
#pragma unroll
                for (int r = 0; r < 8; r++) {
                    float ev = __expf(acc[r]);
                    se += wav[r] * ev;
                    st += kv[r] * (eav[r] * ev);
                }
                edd += wbj * se;
                trc += ebj * st;
            }
        }
    } else {
        for (int ti = wave; ti < 32; ti += 8) {
            int arow = ti * 16 + l16;
            v2f av0, av1, av2;
            av0.x = sPaS[arow * NEP + 2 * lhalf];     av0.y = sPaS[arow * NEP + 2 * lhalf + 1];
            av1.x = sPaS[arow * NEP + 4 + 2 * lhalf]; av1.y = sPaS[arow * NEP + 4 + 2 * lhalf + 1];
            av2.x = sPaS[arow * NEP + 8 + 2 * lhalf]; av2.y = sPaS[arow * NEP + 8 + 2 * lhalf + 1];
            int i0 = ti * 16 + 8 * lhalf;
            float4 wa0 = *(const float4*)&sWa[i0];
            float4 wa1 = *(const float4*)&sWa[i0 + 4];
            float wav[8] = {wa0.x, wa0.y, wa0.z, wa0.w, wa1.x, wa1.y, wa1.z, wa1.w};
            for (int tj = 0; tj < 32; tj++) {
                int bcol = tj * 16 + l16;
                v2f bv0, bv1, bv2;
                bv0.x = sPd[bcol * NEP + 2 * lhalf];     bv0.y = sPd[bcol * NEP + 2 * lhalf + 1];
                bv1.x = sPd[bcol * NEP + 4 + 2 * lhalf]; bv1.y = sPd[bcol * NEP + 4 + 2 * lhalf + 1];
                bv2.x = sPd[bcol * NEP + 8 + 2 * lhalf]; bv2.y = sPd[bcol * NEP + 8 + 2 * lhalf + 1];
                v8f acc = {};
                acc = __builtin_amdgcn_wmma_f32_16x16x4_f32(false, av0, false, bv0, (short)0, acc, false, false);
                acc = __builtin_amdgcn_wmma_f32_16x16x4_f32(false, av1, false, bv1, (short)0, acc, false, false);
                acc = __builtin_amdgcn_wmma_f32_16x16x4_f32(false, av2, false, bv2, (short)0, acc, false, false);
                float wbj = sWb[bcol];
                float se = 0.f;
#pragma unroll
                for (int r = 0; r < 8; r++) {
                    float ev = __expf(acc[r]);
                    se += wav[r] * ev;
                }
                edd += wbj * se;
            }
        }
    }

    // wave32 reduction then cross-wave via LDS
    for (int off = 16; off > 0; off >>= 1) {
        edd += __shfl_xor(edd, off, 32);
        trc += __shfl_xor(trc, off, 32);
    }
    if (lane == 0) { sRed[wave] = edd; sRed[8 + wave] = trc; }
    __syncthreads();
    if (tid == 0) {
        float se = 0.f, st = 0.f;
        for (int w = 0; w < 8; w++) { se += sRed[w]; st += sRed[8 + w]; }
        ws[OFF_EDD + (size_t)(b * ND + a) * ND + d] = se;
        if (a == d) ws[OFF_TR + b * ND + a] = st;
    }
}

// ====== kernel 6: finalize mu_next [B,D] and Sigma_next [B,D,D] ======
__global__ void k_final(const float* __restrict__ obs_mean,
                        const float* __restrict__ obs_var,
                        const float* __restrict__ alpha_sq,
                        const float* __restrict__ sig_eps,
                        const float* __restrict__ ws,
                        float* __restrict__ out) {
    const float* MUD = ws + OFF_MUD;
    const float* EDD = ws + OFF_EDD;
    const float* TR  = ws + OFF_TR;
    const float* V   = ws + OFF_V;
    for (int idx = threadIdx.x; idx < NB * ND + NB * ND * ND; idx += blockDim.x) {
        if (idx < NB * ND) {
            out[idx] = obs_mean[idx] + MUD[idx];
        } else {
            int r = idx - NB * ND;
            int b = r / (ND * ND);
            int i = (r / ND) % ND;
            int j = r % ND;
            float eij = EDD[(b * ND + i) * ND + j];
            float eji = EDD[(b * ND + j) * ND + i];
            float sd = 0.5f * (eij + eji) - MUD[b * ND + i] * MUD[b * ND + j];
            if (i == j) sd += alpha_sq[i] - TR[b * ND + i] + sig_eps[i];
            float cxd_ij = V[(size_t)(b * ND + j) * NE + i];  // C_xd[b,i,j] = V[b,j,i]
            float cxd_ji = V[(size_t)(b * ND + i) * NE + j];
            out[idx] = obs_var[r] + sd + cxd_ij + cxd_ji;
        }
    }
}

extern "C" void kernel_launch(void* const* d_in, const int* in_sizes, int n_in,
                              void* d_out, int out_size, void* d_ws, size_t ws_size,
                              hipStream_t stream) {
    const float* obs_mean = (const float*)d_in[0];
    const float* obs_var  = (const float*)d_in[1];
    const float* act_mean = (const float*)d_in[2];
    const float* act_var  = (const float*)d_in[3];
    const float* cross    = (const float*)d_in[4];
    const float* X        = (const float*)d_in[5];
    const float* ell      = (const float*)d_in[6];
    const float* alpha_sq = (const float*)d_in[7];
    const float* sig_eps  = (const float*)d_in[8];
    const float* beta     = (const float*)d_in[9];
    const float* invK     = (const float*)d_in[10];
    float* ws  = (float*)d_ws;
    float* out = (float*)d_out;

    k_assemble<<<1, 256, 0, stream>>>(obs_mean, obs_var, act_mean, act_var, cross, ws);
    k_ainv<<<NB * ND, 32, 0, stream>>>(ell, ws);
    k_ssolve<<<ND * NB * ND, 32, 0, stream>>>(ell, ws);
    k_qkern<<<NB * ND, 256, 0, stream>>>(X, ell, alpha_sq, beta, ws);
    k_heavy<<<ND * NB * ND, 256, 0, stream>>>(X, ell, beta, invK, ws);
    k_final<<<1, 256, 0, stream>>>(obs_mean, obs_var, alpha_sq, sig_eps, ws, out);
}